// CodeBook_6090263625898
// MI455X (gfx1250) — compile-verified
//
#include <hip/hip_runtime.h>

#define N_ROWS 16384
#define H_DIM  2048
#define M_CODES 64
#define KCHUNK 128
#define NCHUNK (H_DIM / KCHUNK)

typedef __attribute__((ext_vector_type(2))) float v2f;
typedef __attribute__((ext_vector_type(8))) float v8f;

typedef unsigned int uint32x4 __attribute__((ext_vector_type(4)));
typedef int          int32x4  __attribute__((ext_vector_type(4)));
typedef int          int32x8  __attribute__((ext_vector_type(8)));

#ifdef __has_builtin
#  if __has_builtin(__builtin_amdgcn_tensor_load_to_lds)
#    define HAVE_TDM 1
#  endif
#endif
#ifndef HAVE_TDM
#  define HAVE_TDM 0
#endif

#if HAVE_TDM
static __device__ __forceinline__ void wait_tensor0() {
#  if __has_builtin(__builtin_amdgcn_s_wait_tensorcnt)
  __builtin_amdgcn_s_wait_tensorcnt(0);
#  else
  asm volatile("s_wait_tensorcnt 0x0" ::: "memory");
#  endif
}

// Issue one TDM DMA: 64x128 f32 tile of w (tile start = &w[kc]) -> LDS offset.
// D# per CDNA5 ISA 8.3/8.4: 2D tensor dim0=H (contiguous), dim1=64 rows,
// dim0_stride=H, tile 128x64, data_size=4B, group0.type=2.
// This toolchain's builtin takes 6 args: (g0, g1, g2, g3, g4, cpol).
static __device__ __forceinline__ void tdm_load_w_chunk(const float* w, int kc,
                                                        unsigned lds_off) {
  unsigned long long ga = (unsigned long long)(const void*)(w + kc);
  uint32x4 g0 = {};
  g0.x = 1u;                                                   // count=1 (valid)
  g0.y = lds_off;                                              // lds_addr (bytes)
  g0.z = (unsigned)(ga & 0xffffffffull);                       // global_addr[31:0]
  g0.w = (unsigned)((ga >> 32) & 0x1ffffffull) | (2u << 30);   // [56:32] | type=2
  int32x8 g1 = {};
  g1[0] = (int)(2u << 16);                 // wg_mask=0 | data_size=2 (4 bytes)
  g1[1] = (int)((unsigned)H_DIM << 16);    // tensor_dim0[15:0] in [31:16]
  g1[2] = (int)((unsigned)M_CODES << 16);  // tensor_dim0[31:16]=0 | tensor_dim1 lo16
  g1[3] = (int)((unsigned)KCHUNK << 16);   // tensor_dim1 hi16=0 | tile_dim0=128
  g1[4] = M_CODES;                         // tile_dim1=64, tile_dim2=0
  g1[5] = H_DIM;                           // tensor_dim0_stride[31:0]
  g1[6] = 0;                               // stride hi | dim1_stride lo (unused, 2D)
  g1[7] = 0;
  int32x4 g2 = {};                         // groups 2/3 unused for 2D tile
  int32x4 g3 = {};
  int32x8 g4 = {};                         // trailing group (6-arg form), unused
  __builtin_amdgcn_tensor_load_to_lds(g0, g1, g2, g3, g4, 0);
}

static __device__ __forceinline__ unsigned lds_byte_off(const void* p) {
  // Generic pointers to LDS carry the LDS byte offset in the low 32 bits.
  return (unsigned)(unsigned long long)p;
}
#endif

// ---------------------------------------------------------------------------
// Precompute ||w_m||^2 for the 64 codewords (w is tiny / L2-resident).
// Keeps the WMMA hot loop free of B-fragment reuse -> no register copies.
// ---------------------------------------------------------------------------
__global__ __launch_bounds__(256) void wnorm_kernel(const float* __restrict__ w,
                                                    float* __restrict__ wn2) {
  const int mrow  = blockIdx.x;
  const float* wr = w + (size_t)mrow * H_DIM;
  float s = 0.f;
  for (int h = threadIdx.x; h < H_DIM; h += 256) {
    const float q = wr[h];
    s += q * q;
  }
  __shared__ float sd[256];
  sd[threadIdx.x] = s;
  __syncthreads();
  for (int st = 128; st > 0; st >>= 1) {
    if (threadIdx.x < st) sd[threadIdx.x] += sd[threadIdx.x + st];
    __syncthreads();
  }
  if (threadIdx.x == 0) wn2[mrow] = sd[0];
}

// ---------------------------------------------------------------------------
// Kernel A: D[r][m] = sqrt(||x_r||^2 + ||w_m||^2 - 2 x_r.w_m) via f32 WMMA.
// 256 threads = 8 waves; each wave computes a 16-row x 64-col slab.
// w is staged through LDS in double-buffered 128-wide K chunks via the
// Tensor Data Mover (async DMA, TENSORcnt), overlapping DMA with WMMA.
// Row squared norms are accumulated in place from the A fragments.
// ---------------------------------------------------------------------------
__global__ __launch_bounds__(256) void dist_wmma_kernel(const float* __restrict__ x,
                                                        const float* __restrict__ w,
                                                        const float* __restrict__ wn2,
                                                        float* __restrict__ dist) {
  __shared__ float lw[2][M_CODES * KCHUNK];   // 2 x 32 KB chunks of w
  __shared__ float l_xs[8][32];

  const int tid  = threadIdx.x;
  const int wv   = tid >> 5;
  const int lane = tid & 31;
  const int m    = lane & 15;
  const int half = lane >> 4;
  const int row0 = blockIdx.x * 128 + wv * 16;

  v8f   c[4] = {};
  float xs   = 0.f;

#if HAVE_TDM
  if (tid < 32) tdm_load_w_chunk(w, 0, lds_byte_off(&lw[0][0]));   // prologue DMA
#endif

  for (int ci = 0; ci < NCHUNK; ++ci) {
    const int cur = ci & 1;
    const int kc  = ci * KCHUNK;
#if HAVE_TDM
    wait_tensor0();        // chunk `ci` DMA complete (no-op for non-issuing waves)
    __syncthreads();       // LDS visible to all; everyone done reading buf cur^1
    if (ci + 1 < NCHUNK && tid < 32)
      tdm_load_w_chunk(w, kc + KCHUNK, lds_byte_off(&lw[cur ^ 1][0]));
#else
    __syncthreads();
    for (int e = tid; e < M_CODES * KCHUNK; e += 256) {
      lw[cur][e] = w[(size_t)(e >> 7) * H_DIM + kc + (e & 127)];
    }
    __syncthreads();
#endif

    const float* lwb  = lw[cur];
    const float* xrow = x + (size_t)(row0 + m) * H_DIM + kc + 2 * half;
    for (int kk = 0; kk < KCHUNK; kk += 4) {
      // A fragment: 16x4 f32, lanes 0-15 hold K={0,1}, lanes 16-31 K={2,3}
      v2f a = *(const v2f*)(xrow + kk);
      xs += a.x * a.x + a.y * a.y;
#pragma unroll
      for (int t = 0; t < 4; ++t) {
        const int n = t * 16 + m;
        // B fragment: 4x16 f32, N across lanes, K pair selected by lane half
        v2f b = *(const v2f*)(&lwb[n * KCHUNK + kk + 2 * half]);
        c[t] = __builtin_amdgcn_wmma_f32_16x16x4_f32(false, a, false, b,
                                                     (short)0, c[t], false, false);
      }
    }
  }

  __syncthreads();
  l_xs[wv][lane] = xs;
  __syncthreads();

#pragma unroll
  for (int t = 0; t < 4; ++t) {
    const float wn2v = wn2[t * 16 + m];
#pragma unroll
    for (int e = 0; e < 8; ++e) {
      const int   rl  = e + 8 * half;                     // local row (C/D layout)
      const float xn2 = l_xs[wv][rl] + l_xs[wv][rl + 16];
      const float d2  = xn2 + wn2v - 2.0f * c[t][e];
      dist[(size_t)(row0 + rl) * M_CODES + t * 16 + m] = sqrtf(fmaxf(d2, 0.f));
    }
  }
}

// ---------------------------------------------------------------------------
// Kernel B: Jonker-Volgenant LSA on each 64x64 block. One workgroup (64
// threads) per block; relaxation + argmin parallel, augmenting walk serial.
// ---------------------------------------------------------------------------
__global__ __launch_bounds__(64) void lsa_kernel(const float* __restrict__ dist,
                                                 int* __restrict__ idx_out) {
  const int b = blockIdx.x;
  const int t = threadIdx.x;
  const float INFV = 1e30f;

  __shared__ float C[64 * 64];
  __shared__ float u[65], v[65], minv[65];
  __shared__ int   p[65], way[65], used[65];
  __shared__ float rval[64];
  __shared__ int   ridx[64];
  __shared__ int   s_j0;

  const float* Db = dist + (size_t)b * 64 * 64;
  for (int e = t; e < 64 * 64; e += 64) C[e] = Db[e];
  u[t] = 0.f; v[t] = 0.f; p[t] = 0; way[t] = 0;
  if (t == 0) { u[64] = 0.f; v[64] = 0.f; p[64] = 0; way[64] = 0; }
  __syncthreads();

  for (int i = 1; i <= 64; ++i) {
    if (t == 0) { p[0] = i; s_j0 = 0; used[0] = 0; }
    minv[t + 1] = INFV;
    used[t + 1] = 0;
    __syncthreads();

    while (true) {
      if (t == 0) used[s_j0] = 1;
      __syncthreads();
      const int j0 = s_j0;
      const int i0 = p[j0];
      const int j  = t + 1;
      float mv = minv[j];
      if (!used[j]) {
        const float cur = C[(i0 - 1) * 64 + (j - 1)] - u[i0] - v[j];
        if (cur < mv) { mv = cur; minv[j] = cur; way[j] = j0; }
      }
      rval[t] = used[j] ? INFV : mv;
      ridx[t] = j;
      __syncthreads();
      for (int s = 32; s > 0; s >>= 1) {
        if (t < s && rval[t + s] < rval[t]) { rval[t] = rval[t + s]; ridx[t] = ridx[t + s]; }
        __syncthreads();
      }
      const int   j1    = ridx[0];
      const float delta = rval[0];

      // dual updates: thread t handles index t, thread 0 also index 64
      if (used[t]) { u[p[t]] += delta; v[t] -= delta; }
      else if (t > 0) minv[t] -= delta;
      if (t == 0) {
        if (used[64]) { u[p[64]] += delta; v[64] -= delta; }
        else minv[64] -= delta;
        s_j0 = j1;
      }
      const bool done = (p[j1] == 0);  // read before barrier releases augment writes
      __syncthreads();
      if (done) break;
    }

    if (t == 0) {  // augmenting path walk (serial)
      int j0 = s_j0;
      while (j0) {
        const int jp = way[j0];
        p[j0] = p[jp];
        j0 = jp;
      }
    }
    __syncthreads();
  }

  const int j = t + 1;
  idx_out[(size_t)b * 64 + (p[j] - 1)] = j - 1;
}

// ---------------------------------------------------------------------------
// Loss reductions + gather
// ---------------------------------------------------------------------------
__global__ void zero_acc_kernel(float* acc) { acc[0] = 0.f; }

__global__ __launch_bounds__(256) void gather_mse_kernel(const float* __restrict__ x,
                                                         const float* __restrict__ w,
                                                         const int* __restrict__ idx,
                                                         float* __restrict__ outq,
                                                         float* __restrict__ acc) {
  const int n    = blockIdx.x;
  const int code = idx[n];
  const float* wr = w + (size_t)code * H_DIM;
  const float* xr = x + (size_t)n * H_DIM;
  float*       qr = outq + (size_t)n * H_DIM;
  float s = 0.f;
  for (int h = threadIdx.x; h < H_DIM; h += 256) {
    const float q = wr[h];
    const float d = xr[h] - q;
    s += d * d;
    qr[h] = q;
  }
  __shared__ float sd[256];
  sd[threadIdx.x] = s;
  __syncthreads();
  for (int st = 128; st > 0; st >>= 1) {
    if (threadIdx.x < st) sd[threadIdx.x] += sd[threadIdx.x + st];
    __syncthreads();
  }
  if (threadIdx.x == 0) atomicAdd(acc, sd[0]);
}

__global__ __launch_bounds__(256) void colsum_partial_kernel(const float* __restrict__ x,
                                                             float* __restrict__ colpart) {
  const int h = blockIdx.x * 256 + threadIdx.x;
  const int y = blockIdx.y;                 // 16 row slabs of 1024 rows
  float s = 0.f;
  for (int n = y * 1024; n < (y + 1) * 1024; ++n) s += x[(size_t)n * H_DIM + h];
  colpart[y * H_DIM + h] = s;
}

__global__ __launch_bounds__(256) void colcombine_kernel(const float* __restrict__ w,
                                                         const float* __restrict__ colpart,
                                                         float* __restrict__ col_d2) {
  const int h = blockIdx.x * 256 + threadIdx.x;
  float xs = 0.f;
  for (int yy = 0; yy < 16; ++yy) xs += colpart[yy * H_DIM + h];
  float wsum = 0.f;
  for (int n = 0; n < M_CODES; ++n) wsum += w[(size_t)n * H_DIM + h];
  const float d = xs / (float)N_ROWS - wsum / (float)M_CODES;
  col_d2[h] = d * d;
}

__global__ __launch_bounds__(256) void finalize_kernel(const float* __restrict__ col_d2,
                                                       const float* __restrict__ acc,
                                                       float* __restrict__ loss) {
  __shared__ float sd[256];
  float s = 0.f;
  for (int h = threadIdx.x; h < H_DIM; h += 256) s += col_d2[h];
  sd[threadIdx.x] = s;
  __syncthreads();
  for (int st = 128; st > 0; st >>= 1) {
    if (threadIdx.x < st) sd[threadIdx.x] += sd[threadIdx.x + st];
    __syncthreads();
  }
  if (threadIdx.x == 0) {
    const float mse1 = acc[0] / ((float)N_ROWS * (float)H_DIM);
    const float mse2 = sd[0] / (float)H_DIM;
    loss[0] = 1.25f * (mse1 + mse2);
  }
}

// ---------------------------------------------------------------------------
extern "C" void kernel_launch(void* const* d_in, const int* in_sizes, int n_in,
                              void* d_out, int out_size, void* d_ws, size_t ws_size,
                              hipStream_t stream) {
  const float* x = (const float*)d_in[0];   // mode_emb (16384, 2048) f32
  const float* w = (const float*)d_in[1];   // embedding_weight (64, 2048) f32

  float* out     = (float*)d_out;
  float* loss    = out;                                       // [0]
  float* quant   = out + 1;                                   // [1 .. 1+N*H)
  int*   indices = (int*)(out + 1 + (size_t)N_ROWS * H_DIM);  // tail, int32

  float* ws_f    = (float*)d_ws;
  float* col_d2  = ws_f;               // 2048 floats
  float* acc     = ws_f + 2048;        // 1 float
  float* wn2     = ws_f + 2304;        // 64 floats
  float* colpart = ws_f + 4096;        // 16*2048 floats

  // Park the 16384x64 distance matrix inside the quantized output region;
  // gather_mse_kernel overwrites it afterwards (stream-ordered).
  float* dist = quant;

  wnorm_kernel<<<M_CODES, 256, 0, stream>>>(w, wn2);
  dist_wmma_kernel<<<N_ROWS / 128, 256, 0, stream>>>(x, w, wn2, dist);
  lsa_kernel<<<N_ROWS / 64, 64, 0, stream>>>(dist, indices);
  zero_acc_kernel<<<1, 1, 0, stream>>>(acc);
  colsum_partial_kernel<<<dim3(H_DIM / 256, 16), 256, 0, stream>>>(x, colpart);
  gather_mse_kernel<<<N_ROWS, 256, 0, stream>>>(x, w, indices, quant, acc);
  colcombine_kernel<<<H_DIM / 256, 256, 0, stream>>>(w, colpart, col_d2);
  finalize_kernel<<<1, 256, 0, stream>>>(col_d2, acc, loss);
}